// QRNN3DLayer_40243843563912
// MI455X (gfx1250) — compile-verified
//
#include <hip/hip_runtime.h>

typedef float v2f __attribute__((ext_vector_type(2)));
typedef float v8f __attribute__((ext_vector_type(8)));
typedef int   v2i __attribute__((ext_vector_type(2)));

#define B_    4
#define CIN   16
#define HID   16
#define T_    31
#define H_    128
#define W_    128
#define NOC   64             // 4*HID output channels
#define KTOT  432            // CIN*3*3*3, enumerated as (kd, ic, kh, kw)
#define KPD   144            // K per kd sub-loop: CIN*3*3
#define XS_W  130            // 128 + w halo
#define PLANE_ELEMS (CIN*4*XS_W)      // one t-plane: [ic][4 h-rows][130] = 8320
#define XS_ELEMS (4*PLANE_ELEMS)      // 4-slot circular buffer = 33280
#define BS_ELEMS (KTOT*NOC)           // 27648 (pair-interleaved [K/2][N][2])
#define HS_ELEMS (HID*2*W_)           // 4096 (2 output rows)
#define OFF_N   KPD
#define SMEM_BYTES ((BS_ELEMS + XS_ELEMS + HS_ELEMS) * 4 + OFF_N * 4)

__device__ __forceinline__ float fsigmoid(float x) {
    return __frcp_rn(1.f + __expf(-x));
}
__device__ __forceinline__ float ftanh(float x) {
    float e = __expf(-2.f * x);
    return __fdividef(1.f - e, 1.f + e);
}

__global__ __launch_bounds__(256)
void qrnn3d_fused_wmma(const float* __restrict__ X,
                       const float* __restrict__ Wt,
                       const float* __restrict__ bias,
                       float* __restrict__ out)
{
    extern __shared__ float smem[];
    float* Bs = smem;                        // [KTOT/2][NOC][2] pair-interleaved weights
    float* Xs = Bs + BS_ELEMS;               // [4 slots][CIN][4 rows][130] rolling planes
    float* Hs = Xs + XS_ELEMS;               // [HID][2][W] output staging
    int*   Off = (int*)(Hs + HS_ELEMS);      // [KPD] (ic,kh,kw) -> in-plane float offset

    const int tid  = threadIdx.x;
    const int lane = tid & 31;               // wave32
    const int wave = tid >> 5;               // 0..7 == w-tile (16 positions each)
    const int hrow0 = blockIdx.x * 2;        // this block owns h rows {hrow0, hrow0+1}
    const int b     = blockIdx.y;
    const float* Xb = X + (long)b * CIN * T_ * H_ * W_;   // uniform batch base (SGPR)

    // ---- stage weights once, K reordered to (kd, ic, kh, kw) and pair-interleaved
    // so a lane's (k,k+1) fragment is one aligned ds_load_b64 ----
    for (int i = tid; i < BS_ELEMS; i += 256) {     // i = n*KTOT + korig (coalesced read)
        int n = i / KTOT, ko = i % KTOT;
        int ic = ko / 27, r = ko % 27;
        int kd = r / 9, kh = (r % 9) / 3, kw = r % 3;
        int k  = kd * KPD + ic * 9 + kh * 3 + kw;   // our enumeration
        Bs[(k >> 1) * (NOC * 2) + n * 2 + (k & 1)] = Wt[i];
    }
    // ---- in-plane offset table: j = ic*9 + kh*3 + kw ----
    for (int j = tid; j < KPD; j += 256) {
        int ic = j / 9, kh = (j % 9) / 3, kw = j % 3;
        Off[j] = (ic * 4 + kh) * XS_W + kw;
    }
    // ---- prologue: stage planes tt = -1 (zeros), 0, 1 into slots 3, 0, 1 ----
    for (int idx = tid; idx < 3 * PLANE_ELEMS; idx += 256) {
        int p  = idx / PLANE_ELEMS;          // 0..2 -> tt = p-1
        int i  = idx % PLANE_ELEMS;
        int tt = p - 1;
        int ww = i % XS_W, row = i / XS_W;
        int rr = row & 3,  ic  = row >> 2;
        int hh = hrow0 + rr - 1, wg = ww - 1;
        float v = 0.f;
        if (tt >= 0 && (unsigned)hh < (unsigned)H_ && (unsigned)wg < (unsigned)W_)
            v = Xb[((long)ic * T_ + tt) * H_ * W_ + hh * W_ + wg];
        Xs[(tt & 3) * PLANE_ELEMS + i] = v;
    }

    const int ncol = lane & 15;              // N (channel-within-gate) this lane owns
    const float bz = bias[0 * HID + ncol];
    const float bf = bias[1 * HID + ncol];
    const float bo = bias[2 * HID + ncol];
    const float bi = bias[3 * HID + ncol];

    const int mcol  = wave * 16 + ncol;      // w position this lane feeds into A
    const int khalf = (lane < 16) ? 0 : 2;   // 16x16x4 f32 half-wave K split
    const float* bbase = Bs + (khalf >> 1) * (NOC * 2) + ncol * 2;

    float c0[8], c1[8];
    #pragma unroll
    for (int j = 0; j < 8; ++j) { c0[j] = 0.f; c1[j] = 0.f; }

    for (int t = 0; t < T_; ++t) {
        __syncthreads();   // publish prologue / previous async plane; Hs readers done

        // ---- fire async staging of plane t+2 into slot (t+2)&3; it lands while
        // the k-loop below runs. Halo / out-of-range elements get LDS zero-stores.
        {
            const int tt = t + 2;
            float* Xp = Xs + (tt & 3) * PLANE_ELEMS;
            const bool pv = (tt < T_);
            for (int i = tid; i < PLANE_ELEMS; i += 256) {
                int ww = i % XS_W, row = i / XS_W;
                int rr = row & 3,  ic  = row >> 2;
                int hh = hrow0 + rr - 1, wg = ww - 1;
                if (pv && (unsigned)hh < (unsigned)H_ && (unsigned)wg < (unsigned)W_) {
                    unsigned ldsa = (unsigned)(size_t)(Xp + i);
                    unsigned goff = (unsigned)(((ic * T_ + tt) * H_ + hh) * W_ + wg) * 4u;
                    asm volatile("global_load_async_to_lds_b32 %0, %1, %2"
                                 :: "v"(ldsa), "v"(goff), "s"(Xb) : "memory");
                } else {
                    Xp[i] = 0.f;
                }
            }
        }

        // ---- implicit GEMM: gates[2 x 128 x 64] = A[2x128 x 432] * B[432 x 64]
        // K split into 3 kd sub-loops; plane base is uniform per sub-loop ----
        v8f az0 = {}, af0 = {}, ao0 = {}, ai0 = {};
        v8f az1 = {}, af1 = {}, ao1 = {}, ai1 = {};
        #pragma unroll
        for (int kd = 0; kd < 3; ++kd) {
            const float* Xp = Xs + ((t + kd - 1) & 3) * PLANE_ELEMS;
            const float* bkd = bbase + (kd * (KPD / 2)) * (NOC * 2);
            for (int j0 = 0; j0 < KPD; j0 += 4) {
                v2i offs = *(const v2i*)(Off + j0 + khalf);   // one ds_load_b64
                const int a0 = offs.x + mcol;
                const int a1 = offs.y + mcol;
                v2f aA, aB;                                   // row0 / row1 A fragments
                aA.x = Xp[a0];          aA.y = Xp[a1];
                aB.x = Xp[a0 + XS_W];   aB.y = Xp[a1 + XS_W]; // +130 folds to imm offset
                const float* bp = bkd + (j0 >> 1) * (NOC * 2);
                v2f b0 = *(const v2f*)(bp);                   // aligned ds_load_b64 each
                v2f b1 = *(const v2f*)(bp + 32);
                v2f b2 = *(const v2f*)(bp + 64);
                v2f b3 = *(const v2f*)(bp + 96);
                az0 = __builtin_amdgcn_wmma_f32_16x16x4_f32(false, aA, false, b0, (short)0, az0, false, false);
                af0 = __builtin_amdgcn_wmma_f32_16x16x4_f32(false, aA, false, b1, (short)0, af0, false, false);
                ao0 = __builtin_amdgcn_wmma_f32_16x16x4_f32(false, aA, false, b2, (short)0, ao0, false, false);
                ai0 = __builtin_amdgcn_wmma_f32_16x16x4_f32(false, aA, false, b3, (short)0, ai0, false, false);
                az1 = __builtin_amdgcn_wmma_f32_16x16x4_f32(false, aB, false, b0, (short)0, az1, false, false);
                af1 = __builtin_amdgcn_wmma_f32_16x16x4_f32(false, aB, false, b1, (short)0, af1, false, false);
                ao1 = __builtin_amdgcn_wmma_f32_16x16x4_f32(false, aB, false, b2, (short)0, ao1, false, false);
                ai1 = __builtin_amdgcn_wmma_f32_16x16x4_f32(false, aB, false, b3, (short)0, ai1, false, false);
            }
        }

        // ---- activations + sequential cell update (per-lane registers) ----
        // C/D layout: elem j, lanes 0-15: M=j ; lanes 16-31: M=j+8 ; N=lane&15
        const int mrow = wave * 16 + ((lane < 16) ? 0 : 8);
        #pragma unroll
        for (int j = 0; j < 8; ++j) {
            float z  = ftanh(az0[j] + bz);
            float f  = fsigmoid(af0[j] + bf);
            float o  = fsigmoid(ao0[j] + bo);
            float ig = fsigmoid(ai0[j] + bi);
            float cn = f * c0[j] + ig * z;
            c0[j] = cn;
            Hs[(ncol * 2 + 0) * W_ + mrow + j] = o * cn;

            z  = ftanh(az1[j] + bz);
            f  = fsigmoid(af1[j] + bf);
            o  = fsigmoid(ao1[j] + bo);
            ig = fsigmoid(ai1[j] + bi);
            cn = f * c1[j] + ig * z;
            c1[j] = cn;
            Hs[(ncol * 2 + 1) * W_ + mrow + j] = o * cn;
        }

        asm volatile("s_wait_asynccnt 0x0" ::: "memory");  // plane t+2 landed
        __syncthreads();                                   // publish Hs + plane t+2

        // ---- coalesced writeback: out[b][n][t][hrow0+r][w] ----
        for (int i = tid; i < HS_ELEMS; i += 256) {
            int n = i >> 8;
            int r = (i >> 7) & 1;
            int w = i & 127;
            out[((((long)b * HID + n) * T_ + t) * H_ + (hrow0 + r)) * W_ + w] = Hs[i];
        }
    }
}

extern "C" void kernel_launch(void* const* d_in, const int* in_sizes, int n_in,
                              void* d_out, int out_size, void* d_ws, size_t ws_size,
                              hipStream_t stream) {
    const float* X    = (const float*)d_in[0];
    const float* Wt   = (const float*)d_in[1];
    const float* bias = (const float*)d_in[2];
    float* out        = (float*)d_out;
    (void)in_sizes; (void)n_in; (void)out_size; (void)d_ws; (void)ws_size;

    dim3 grid(H_ / 2, B_);           // 64 row-pairs x 4 batches = 256 workgroups
    dim3 block(256);                 // 8 wave32, each owns 16 w-positions x 2 rows
    qrnn3d_fused_wmma<<<grid, block, SMEM_BYTES, stream>>>(X, Wt, bias, out);
}